// LSQ_2448131359011
// MI455X (gfx1250) — compile-verified
//
#include <hip/hip_runtime.h>
#include <hip/hip_bf16.h>

typedef __attribute__((ext_vector_type(2))) float v2f;
typedef __attribute__((ext_vector_type(8))) float v8f;

#define MCB 8
#define KCB 256
#define DD  256
#define CBN (MCB * KCB)   // 2048 codewords total

// ---------------------------------------------------------------------------
// GEMM-NT: C[m,n] = scale * sum_k A[m,k] * B[n,k]
// A: (Ma x 256) row-major, B: (Nb x 256) row-major, C: row-major ldc = 2048.
// Wave32 computes a 16x64 strip of C via 4 accumulators of
// V_WMMA_F32_16X16X4_F32 (f32 kept to match reference argmin semantics;
// the GEMMs are HBM/L2-bound, not FLOP-bound, so lower precision buys nothing).
//
// f32 WMMA fragment layout (CDNA5 ISA 7.12.2):
//   A 16x4 : lanes 0-15 = rows M0..15 holding K={0,1}; lanes 16-31 hold K={2,3}
//   B 4x16 : lanes 0-15 = cols N0..15 holding K={0,1}; lanes 16-31 hold K={2,3}
//   Since B_mat[k][n] = B[n][k] (NT), both fragments are float2 loads at
//   (row = base + lane&15, col = k0 + 2*(lane>>4)).
// ---------------------------------------------------------------------------
__global__ __launch_bounds__(256) void lsq_gemm_nt(const float* __restrict__ A,
                                                   const float* __restrict__ B,
                                                   float* __restrict__ C,
                                                   float scale)
{
    const int lane = threadIdx.x & 31;
    const int wv   = threadIdx.x >> 5;            // 8 waves / block
    const int row0 = blockIdx.y * 16;             // C-tile rows
    const int col0 = blockIdx.x * 512 + wv * 64;  // C-tile cols (4 x 16-wide tiles)

    const int fr = lane & 15;
    const int fk = (lane >> 4) << 1;              // 0 or 2

    const float* Arow = A + (size_t)(row0 + fr) * DD + fk;
    const float* B0   = B + (size_t)(col0 +  0 + fr) * DD + fk;
    const float* B1   = B + (size_t)(col0 + 16 + fr) * DD + fk;
    const float* B2   = B + (size_t)(col0 + 32 + fr) * DD + fk;
    const float* B3   = B + (size_t)(col0 + 48 + fr) * DD + fk;

    v8f acc0 = {}, acc1 = {}, acc2 = {}, acc3 = {};

    for (int k0 = 0; k0 < DD; k0 += 4) {
        v2f a  = *(const v2f*)(Arow + k0);
        v2f b0 = *(const v2f*)(B0 + k0);
        v2f b1 = *(const v2f*)(B1 + k0);
        v2f b2 = *(const v2f*)(B2 + k0);
        v2f b3 = *(const v2f*)(B3 + k0);
        acc0 = __builtin_amdgcn_wmma_f32_16x16x4_f32(false, a, false, b0, (short)0, acc0, false, false);
        acc1 = __builtin_amdgcn_wmma_f32_16x16x4_f32(false, a, false, b1, (short)0, acc1, false, false);
        acc2 = __builtin_amdgcn_wmma_f32_16x16x4_f32(false, a, false, b2, (short)0, acc2, false, false);
        acc3 = __builtin_amdgcn_wmma_f32_16x16x4_f32(false, a, false, b3, (short)0, acc3, false, false);
    }

    // C/D layout: VGPR v -> (M = v + 8*(lane>=16), N = lane&15)
    const int rb = row0 + ((lane >> 4) << 3);
    const int cb = col0 + (lane & 15);
#pragma unroll
    for (int v = 0; v < 8; ++v) {
        const size_t r = (size_t)(rb + v) * CBN;
        C[r + cb +  0] = scale * acc0[v];
        C[r + cb + 16] = scale * acc1[v];
        C[r + cb + 32] = scale * acc2[v];
        C[r + cb + 48] = scale * acc3[v];
    }
}

// cUnary[c] = ||codebook[c]||^2 = 0.5 * cPair[c][c]   (cPair = 2*CB@CB^T)
__global__ void lsq_cunary(const float* __restrict__ cPair, float* __restrict__ cU)
{
    int c = blockIdx.x * blockDim.x + threadIdx.x;
    if (c < CBN) cU[c] = 0.5f * cPair[(size_t)c * CBN + c];
}

// Lower runtime-lane selection from a register array to a v_cndmask chain
// (dynamic VGPR indexing would spill to scratch).
__device__ __forceinline__ int sel8(const int* a, int idx)
{
    int r = a[0];
#pragma unroll
    for (int m = 1; m < MCB; ++m) r = (idx == m) ? a[m] : r;
    return r;
}

// ---------------------------------------------------------------------------
// ICM + accept/reject. One wave32 per point; each lane owns 8 of the 256
// candidates. cPair is symmetric, so pair terms are read along ROWS of cPair
// (contiguous, coalesced) instead of strided columns. cPair (16MB) and x_ci
// (134MB) both reside in the 192MB L2, so all gathers are L2 hits.
// Energy E(b) = sum_i x_ci[n,b_i] + sum_i cUnary[b_i] + 0.5*sum_{i!=j} cPair;
// ||x||^2 cancels in the old-vs-new comparison.
// ---------------------------------------------------------------------------
__global__ __launch_bounds__(256) void lsq_icm(const int* __restrict__ b,
                                               const float* __restrict__ xci,
                                               const float* __restrict__ cU,
                                               const float* __restrict__ cP,
                                               const int* __restrict__ icmIterP,
                                               int* __restrict__ out)
{
    const int lane = threadIdx.x & 31;
    const int wv   = threadIdx.x >> 5;
    const int n    = blockIdx.x * 8 + wv;
    const int iters = *icmIterP;

    const float* xrow = xci + (size_t)n * CBN;

    // absolute codes, replicated across all lanes of the wave
    int cs[MCB], cs0[MCB];
    int myc = (lane < MCB) ? b[(size_t)n * MCB + lane] : 0;
#pragma unroll
    for (int m = 0; m < MCB; ++m) {
        int c = __shfl(myc, m, 32) + m * KCB;
        cs[m] = c;
        cs0[m] = c;
    }

    auto energy = [&](const int* codes) -> float {
        float e = 0.f;
        if (lane < MCB) {
            int c = sel8(codes, lane);
            float p = 0.f;
#pragma unroll
            for (int j = 0; j < MCB; ++j) {
                float v = cP[(size_t)c * CBN + codes[j]];
                p += (j == lane) ? 0.f : v;
            }
            e = xrow[c] + cU[c] + 0.5f * p;
        }
#pragma unroll
        for (int off = 16; off; off >>= 1) e += __shfl_xor(e, off, 32);
        return e;
    };

    const float oldE = energy(cs0);

    for (int it = 0; it < iters; ++it) {
        for (int i = 0; i < MCB; ++i) {
            const float* xs = xrow + i * KCB;
            const float* us = cU + i * KCB;
            float cost[8];
#pragma unroll
            for (int t = 0; t < 8; ++t)
                cost[t] = xs[lane + 32 * t] + us[lane + 32 * t];
#pragma unroll
            for (int m = 0; m < MCB; ++m) {
                if (m == i) continue;  // i is loop-constant after unrolling
                const float* pr = cP + (size_t)cs[m] * CBN + i * KCB;  // symmetric row read
#pragma unroll
                for (int t = 0; t < 8; ++t) cost[t] += pr[lane + 32 * t];
            }
            // local argmin (candidate index = lane + 32*t; scan order keeps first-min)
            float bv = cost[0];
            int   bk = lane;
#pragma unroll
            for (int t = 1; t < 8; ++t) {
                int k = lane + 32 * t;
                if (cost[t] < bv) { bv = cost[t]; bk = k; }
            }
            // cross-lane argmin, ties -> lowest index (matches jnp.argmin)
#pragma unroll
            for (int off = 16; off; off >>= 1) {
                float ov = __shfl_xor(bv, off, 32);
                int   ok = __shfl_xor(bk, off, 32);
                if (ov < bv || (ov == bv && ok < bk)) { bv = ov; bk = ok; }
            }
            cs[i] = i * KCB + bk;
        }
    }

    const float newE = energy(cs);
    const bool worse = (newE >= oldE);
    if (lane < MCB) {
        int oc = sel8(cs0, lane);
        int nc = sel8(cs, lane);
        out[(size_t)n * MCB + lane] = (worse ? oc : nc) - lane * KCB;
    }
}

// ---------------------------------------------------------------------------
extern "C" void kernel_launch(void* const* d_in, const int* in_sizes, int n_in,
                              void* d_out, int out_size, void* d_ws, size_t ws_size,
                              hipStream_t stream)
{
    const float* x   = (const float*)d_in[0];   // (N, 256)
    const int*   b   = (const int*)  d_in[1];   // (N, 8)
    const float* cbk = (const float*)d_in[2];   // (2048, 256)
    // d_in[3] = npert (unused, 0)
    const int* icmIter = (const int*)d_in[4];
    int* out = (int*)d_out;

    const int Npts = in_sizes[0] / DD;          // 16384

    // workspace layout: cPair (16MB) | cUnary (8KB) | x_ci (N*2048*4 = 134MB)
    float* cPair = (float*)d_ws;
    float* cU    = cPair + (size_t)CBN * CBN;
    float* xci   = cU + CBN;

    // cPair = 2 * CB @ CB^T     (2048 x 2048)
    lsq_gemm_nt<<<dim3(CBN / 512, CBN / 16), 256, 0, stream>>>(cbk, cbk, cPair, 2.0f);
    // cUnary from cPair diagonal
    lsq_cunary<<<CBN / 256, 256, 0, stream>>>(cPair, cU);
    // x_ci = -2 * x @ CB^T      (N x 2048)
    lsq_gemm_nt<<<dim3(CBN / 512, Npts / 16), 256, 0, stream>>>(x, cbk, xci, -2.0f);
    // ICM sweeps + accept/reject, one wave per point
    lsq_icm<<<Npts / 8, 256, 0, stream>>>(b, xci, cU, cPair, icmIter, out);
}